// GBAG_25074019074664
// MI455X (gfx1250) — compile-verified
//
#include <hip/hip_runtime.h>
#include <stdint.h>

#define BATCH    8192
#define D_IN     4096
#define D_HID    2048
#define D_OUT    512
#define NNZ1_N   32768
#define NNZ2_N   8192

#define TB       16          // batch rows per workgroup
#define NTHREADS 1024        // 32 waves (wave32)
#define NWAVES   (NTHREADS / 32)
#define CHUNK    2048        // sparse entries staged per TDM chunk

// TDM path only in device compilation and only if the builtin exists.
#if defined(__HIP_DEVICE_COMPILE__) && __has_builtin(__builtin_amdgcn_tensor_load_to_lds) && __has_builtin(__builtin_amdgcn_s_wait_tensorcnt)
#define HAVE_TDM 1
#else
#define HAVE_TDM 0
#endif

typedef unsigned int u32x4 __attribute__((ext_vector_type(4)));
typedef int          i32x8 __attribute__((ext_vector_type(8)));
typedef int          i32x4 __attribute__((ext_vector_type(4)));

#if HAVE_TDM
// 1-D DMA of `nelem` 32-bit words from global memory to LDS via the Tensor
// Data Mover.  Descriptor layout per CDNA5 ISA:
//   group0: count=1, lds_addr, global_addr[56:0], type=2
//   group1: data_size=2 (4B), tensor_dim0=nelem, tensor_dim1=1,
//           tile_dim0=nelem, tile_dim1/2=0 (unused), dim0_stride=nelem
__device__ __forceinline__ void tdm_copy_words(const void* gsrc,
                                               uint32_t lds_byte_off,
                                               uint32_t nelem) {
  uint64_t ga = (uint64_t)(uintptr_t)gsrc;
  u32x4 g0;
  g0.x = 1u;                                    // count=1 (valid descriptor)
  g0.y = lds_byte_off;                          // lds_addr [63:32]
  g0.z = (uint32_t)ga;                          // global_addr[31:0]
  g0.w = ((uint32_t)(ga >> 32) & 0x01FFFFFFu)   // global_addr[56:32]
       | 0x80000000u;                           // type=2 ("image") @ [127:126]
  i32x8 g1;
  g1[0] = (int)(2u << 16);                      // data_size=2 -> 4 bytes
  g1[1] = (int)((nelem & 0xFFFFu) << 16);       // tensor_dim0[15:0] @ bits 63:48
  g1[2] = (int)(((nelem >> 16) & 0xFFFFu)       // tensor_dim0[31:16]
       | (1u << 16));                           // tensor_dim1 = 1
  g1[3] = (int)((nelem & 0xFFFFu) << 16);       // tile_dim0 @ bits 127:112
  g1[4] = 0;                                    // tile_dim1=0, tile_dim2=0 (unused)
  g1[5] = (int)nelem;                           // tensor_dim0_stride[31:0]
  g1[6] = 0;
  g1[7] = 0;
  i32x4 z4; z4[0] = 0; z4[1] = 0; z4[2] = 0; z4[3] = 0;
  i32x8 z8; z8[0] = 0; z8[1] = 0; z8[2] = 0; z8[3] = 0;
           z8[4] = 0; z8[5] = 0; z8[6] = 0; z8[7] = 0;
  // clang-23 / therock signature: (v4u, v8i, v4i, v4i, v8i, i32 cpol)
  __builtin_amdgcn_tensor_load_to_lds(g0, g1, z4, z4, z8, 0);
}
#endif

__global__ __launch_bounds__(NTHREADS, 1)
void fused_sparse_mlp(const float* __restrict__ x,
                      const float* __restrict__ w1,
                      const float* __restrict__ b1,
                      const float* __restrict__ w2,
                      const float* __restrict__ b2,
                      const int*   __restrict__ rows1,
                      const int*   __restrict__ cols1,
                      const int*   __restrict__ rows2,
                      const int*   __restrict__ cols2,
                      float*       __restrict__ out) {
  // hTile[j][bl] at j*16+bl : 16 lanes -> 16 distinct LDS banks (conflict-free)
  __shared__ float hTile[D_HID * TB];   // 128 KB
  __shared__ float oTile[D_OUT * TB];   //  32 KB
#if HAVE_TDM
  __shared__ int stage[2][3 * CHUNK];   //  48 KB : [w | rows | cols] x doublebuf
#endif

  const int tid  = threadIdx.x;
  const int lane = tid & 31;
  const int wave = tid >> 5;
  const int half = lane >> 4;   // each half-wave processes its own entry
  const int bl   = lane & 15;   // batch row within tile
  const int b0   = blockIdx.x * TB;

  // ---- init hTile with bias: hTile[j*16+bl] = b1[j] ----
  for (int idx = tid; idx < D_HID * TB; idx += NTHREADS)
    hTile[idx] = b1[idx >> 4];

#if HAVE_TDM
  if (wave == 0) {
    uint32_t soff = (uint32_t)(uintptr_t)(&stage[0][0]);
    tdm_copy_words(w1,    soff,             CHUNK);
    tdm_copy_words(rows1, soff + 4 * CHUNK, CHUNK);
    tdm_copy_words(cols1, soff + 8 * CHUNK, CHUNK);
  }
#endif
  __syncthreads();

  // =================== layer 1: h += w1[e] * x[:, cols1[e]] ===================
#if HAVE_TDM
  {
    int buf = 0;
    for (int base = 0; base < NNZ1_N; base += CHUNK) {
      if (wave == 0) {
        if (base + CHUNK < NNZ1_N) {   // prefetch next chunk, wait for current
          uint32_t soff = (uint32_t)(uintptr_t)(&stage[buf ^ 1][0]);
          tdm_copy_words(w1    + base + CHUNK, soff,             CHUNK);
          tdm_copy_words(rows1 + base + CHUNK, soff + 4 * CHUNK, CHUNK);
          tdm_copy_words(cols1 + base + CHUNK, soff + 8 * CHUNK, CHUNK);
          __builtin_amdgcn_s_wait_tensorcnt(3);
        } else {
          __builtin_amdgcn_s_wait_tensorcnt(0);
        }
      }
      __syncthreads();  // staged chunk visible to all waves
      volatile const int* sp = &stage[buf][0];   // volatile: TDM writes are
                                                 // invisible to the optimizer
      for (int i = 2 * wave + half; i < CHUNK; i += 2 * NWAVES) {
        float w = __int_as_float(sp[i]);
        int   r = sp[CHUNK + i];
        int   c = sp[2 * CHUNK + i];
        float v = x[(size_t)(b0 + bl) * D_IN + c] * w;
        atomicAdd(&hTile[(r << 4) + bl], v);     // ds_add_f32, bank-conflict-free
      }
      __syncthreads();  // all reads of stage[buf] done before it is re-filled
      buf ^= 1;
    }
  }
#else
  for (int e = 2 * wave + half; e < NNZ1_N; e += 2 * NWAVES) {
    float w = w1[e];
    int   r = rows1[e];
    int   c = cols1[e];
    float v = x[(size_t)(b0 + bl) * D_IN + c] * w;
    atomicAdd(&hTile[(r << 4) + bl], v);
  }
  __syncthreads();
#endif

  // ---- issue layer-2 chunk-0 DMA early, overlap with sigmoid ----
#if HAVE_TDM
  if (wave == 0) {
    uint32_t soff = (uint32_t)(uintptr_t)(&stage[0][0]);
    tdm_copy_words(w2,    soff,             CHUNK);
    tdm_copy_words(rows2, soff + 4 * CHUNK, CHUNK);
    tdm_copy_words(cols2, soff + 8 * CHUNK, CHUNK);
  }
#endif

  // ---- sigmoid in place + init oTile with bias ----
  for (int idx = tid; idx < D_HID * TB; idx += NTHREADS) {
    float v = hTile[idx];
    hTile[idx] = 1.0f / (1.0f + __expf(-v));
  }
  for (int idx = tid; idx < D_OUT * TB; idx += NTHREADS)
    oTile[idx] = b2[idx >> 4];
  __syncthreads();

  // =================== layer 2: out += w2[e] * h[:, cols2[e]] =================
#if HAVE_TDM
  {
    int buf = 0;
    for (int base = 0; base < NNZ2_N; base += CHUNK) {
      if (wave == 0) {
        if (base + CHUNK < NNZ2_N) {
          uint32_t soff = (uint32_t)(uintptr_t)(&stage[buf ^ 1][0]);
          tdm_copy_words(w2    + base + CHUNK, soff,             CHUNK);
          tdm_copy_words(rows2 + base + CHUNK, soff + 4 * CHUNK, CHUNK);
          tdm_copy_words(cols2 + base + CHUNK, soff + 8 * CHUNK, CHUNK);
          __builtin_amdgcn_s_wait_tensorcnt(3);
        } else {
          __builtin_amdgcn_s_wait_tensorcnt(0);
        }
      }
      __syncthreads();
      volatile const int* sp = &stage[buf][0];
      for (int i = 2 * wave + half; i < CHUNK; i += 2 * NWAVES) {
        float w = __int_as_float(sp[i]);
        int   r = sp[CHUNK + i];
        int   c = sp[2 * CHUNK + i];
        float v = hTile[(c << 4) + bl] * w;      // h read straight from LDS
        atomicAdd(&oTile[(r << 4) + bl], v);
      }
      __syncthreads();
      buf ^= 1;
    }
  }
#else
  for (int e = 2 * wave + half; e < NNZ2_N; e += 2 * NWAVES) {
    float w = w2[e];
    int   r = rows2[e];
    int   c = cols2[e];
    float v = hTile[(c << 4) + bl] * w;
    atomicAdd(&oTile[(r << 4) + bl], v);
  }
  __syncthreads();
#endif

  // ---- write out, coalesced over the output dimension ----
  for (int idx = tid; idx < D_OUT * TB; idx += NTHREADS) {
    int blw = idx >> 9;            // idx / D_OUT
    int o   = idx & (D_OUT - 1);   // idx % D_OUT
    out[(size_t)(b0 + blw) * D_OUT + o] = oTile[(o << 4) + blw];
  }
}

extern "C" void kernel_launch(void* const* d_in, const int* in_sizes, int n_in,
                              void* d_out, int out_size, void* d_ws, size_t ws_size,
                              hipStream_t stream) {
  (void)in_sizes; (void)n_in; (void)out_size; (void)d_ws; (void)ws_size;
  const float* x     = (const float*)d_in[0];
  const float* w1    = (const float*)d_in[1];
  const float* b1    = (const float*)d_in[2];
  const float* w2    = (const float*)d_in[3];
  const float* b2    = (const float*)d_in[4];
  const int*   rows1 = (const int*)d_in[5];
  const int*   cols1 = (const int*)d_in[6];
  const int*   rows2 = (const int*)d_in[7];
  const int*   cols2 = (const int*)d_in[8];
  float* out = (float*)d_out;

  dim3 grid(BATCH / TB);
  dim3 block(NTHREADS);
  fused_sparse_mlp<<<grid, block, 0, stream>>>(x, w1, b1, w2, b2,
                                               rows1, cols1, rows2, cols2, out);
}